// CTCGreedyDecoder_53781580480551
// MI455X (gfx1250) — compile-verified
//
#include <hip/hip_runtime.h>

typedef float f4 __attribute__((ext_vector_type(4)));

#define SEQ   2048
#define BATCH 32
#define VOCAB 512

// -------- Phase 1: wave-per-row argmax over vocab --------------------------
// x layout: (SEQ, BATCH, VOCAB) row-major; row r = t*BATCH + b is 512
// contiguous floats (2 KB). One wave32 per row: 4 x b128 NT loads per lane.
__global__ __launch_bounds__(256)
void ctc_argmax_kernel(const float* __restrict__ x, int* __restrict__ ml) {
    const int lane = threadIdx.x & 31;
    const int wave = threadIdx.x >> 5;
    const int row  = blockIdx.x * 8 + wave;          // row = t*BATCH + b
    const f4* rowp = (const f4*)(x + (size_t)row * VOCAB);

    float best = -__builtin_inff();
    int   bidx = 0;
#pragma unroll
    for (int k = 0; k < 4; ++k) {
        // Wave-contiguous 512B per load instruction; single-use data -> NT.
        f4 v = __builtin_nontemporal_load(rowp + k * 32 + lane);
        const int base = k * 128 + lane * 4;
#pragma unroll
        for (int j = 0; j < 4; ++j) {
            // per-lane indices strictly increase, so '>' keeps first occurrence
            float val = v[j];
            if (val > best) { best = val; bidx = base + j; }
        }
    }
    // wave32 butterfly argmax reduction; tie-break to lowest index
    // ((max,min-idx) is associative+commutative, so butterfly is exact)
#pragma unroll
    for (int off = 16; off >= 1; off >>= 1) {
        float ov = __shfl_xor(best, off, 32);
        int   oi = __shfl_xor(bidx, off, 32);
        if (ov > best || (ov == best && oi < bidx)) { best = ov; bidx = oi; }
    }
    if (lane == 0) {
        const int t = row >> 5;          // row / BATCH   (BATCH == 32)
        const int b = row & 31;          // row % BATCH
        ml[b * SEQ + t] = bidx;          // store transposed: (batch, seq)
    }
}

// -------- Phase 2: wave-per-batch stream compaction ------------------------
// keep[t] = (t < len) && (ml[t] != blank) && (ml[t-1]==blank || ml[t]!=ml[t-1])
// with ml[-1] == -1 sentinel. Exclusive scan of keep-counts across lanes,
// then scatter kept tokens left-packed; pad with -1; write out_lengths.
__global__ __launch_bounds__(32)
void ctc_compact_kernel(const int* __restrict__ ml,
                        const int* __restrict__ lengths,
                        const int* __restrict__ blankp,
                        float* __restrict__ out) {
    const int b     = blockIdx.x;
    const int lane  = threadIdx.x;
    const int blank = blankp[0];
    const int len   = lengths[b];
    const int* mlb  = ml  + b * SEQ;
    float*     tok  = out + b * SEQ;

    const int CHUNK = SEQ / 32;          // 64 timesteps per lane
    const int t0    = lane * CHUNK;

    // 1) fill padding (-1) for the whole row (stores precede scatter stores;
    //    same-wave stores complete in order at the coherence point)
#pragma unroll 4
    for (int j = 0; j < CHUNK; ++j) tok[t0 + j] = -1.0f;

    // 2) count kept symbols in this lane's chunk
    const int prev0 = (lane == 0) ? -1 : mlb[t0 - 1];
    int p = prev0, cnt = 0;
    for (int j = 0; j < CHUNK; ++j) {
        const int t = t0 + j;
        const int s = mlb[t];
        const bool keep = (t < len) && (s != blank) && ((p == blank) || (s != p));
        cnt += keep ? 1 : 0;
        p = s;
    }

    // 3) wave32 inclusive scan -> exclusive prefix + total
    int incl = cnt;
#pragma unroll
    for (int off = 1; off < 32; off <<= 1) {
        int v = __shfl_up(incl, off, 32);
        if (lane >= off) incl += v;
    }
    const int excl  = incl - cnt;
    const int total = __shfl(incl, 31, 32);

    // 4) scatter kept symbols left-packed in time order
    p = prev0;
    int pos = excl;
    for (int j = 0; j < CHUNK; ++j) {
        const int t = t0 + j;
        const int s = mlb[t];
        const bool keep = (t < len) && (s != blank) && ((p == blank) || (s != p));
        if (keep) tok[pos++] = (float)s;
        p = s;
    }

    if (lane == 0) out[BATCH * SEQ + b] = (float)total;
}

extern "C" void kernel_launch(void* const* d_in, const int* in_sizes, int n_in,
                              void* d_out, int out_size, void* d_ws, size_t ws_size,
                              hipStream_t stream) {
    const float* x       = (const float*)d_in[0];   // (2048, 32, 512) fp32
    const int*   lengths = (const int*)d_in[1];     // (32,) int32
    const int*   blank   = (const int*)d_in[2];     // scalar int32
    float*       out     = (float*)d_out;           // 32*2048 tokens + 32 lengths
    int*         ml      = (int*)d_ws;              // (32, 2048) argmax indices

    // Phase 1: 65536 rows, 8 waves (256 threads) per block
    ctc_argmax_kernel<<<(SEQ * BATCH) / 8, 256, 0, stream>>>(x, ml);
    // Phase 2: one wave32 per batch
    ctc_compact_kernel<<<BATCH, 32, 0, stream>>>(ml, lengths, blank, out);
}